// GraphSAGE_42408507080744
// MI455X (gfx1250) — compile-verified
//
#include <hip/hip_runtime.h>

// GraphSAGE 2-layer forward for MI455X (gfx1250), fp32 WMMA path.
// N=100000, D=256, MAX_DEG=128, B=1024, fanouts (25,10), OUT=128.
// NOTE: JAX's random column permutation (threefry) cannot be reproduced on
// device; sampling takes the first `fanout` adjacency columns (deterministic).

#define D      256
#define OUTD   128
#define BSZ    1024
#define F1     25
#define F2     10
#define N1     (BSZ * F1)     // 25600
#define MAXDEG 128

typedef float v2f __attribute__((ext_vector_type(2)));
typedef float v4f __attribute__((ext_vector_type(4)));
typedef float v8f __attribute__((ext_vector_type(8)));

// ---------------- sampling: ids1[i*25+j] = adj[ids[i]*128 + j] ----------------
__global__ void k_sample(const int* __restrict__ src_ids, const int* __restrict__ adj,
                         int fanout, int total, int* __restrict__ out_ids) {
    int t = blockIdx.x * blockDim.x + threadIdx.x;
    if (t >= total) return;
    int i = t / fanout, j = t - i * fanout;
    out_ids[t] = adj[(size_t)src_ids[i] * MAXDEG + j];
}

// ------------- neighbor mean straight from adjacency (no ids2 buffer) ---------
// out[row] = mean_j feats[ adj[src_ids[row]*128 + j] ],  j < nn.
// 64 threads/block, one float4 (16B) per lane -> each neighbor row is one
// fully-coalesced 1KB read; feats table (102MB) is L2-resident (192MB L2).
__global__ void k_mean_adj(const float* __restrict__ feats, const int* __restrict__ adj,
                           const int* __restrict__ src_ids, int nn, float inv,
                           float* __restrict__ out) {
    int row = blockIdx.x, col = threadIdx.x << 2;
    const int* nb = adj + (size_t)src_ids[row] * MAXDEG;   // uniform -> s_load
    v4f s = {0.f, 0.f, 0.f, 0.f};
    for (int j = 0; j < nn; ++j)
        s += *(const v4f*)(feats + (size_t)nb[j] * D + col);
    *(v4f*)(out + (size_t)row * D + col) = s * inv;
}

// layer-2 mean: contiguous groups of nn rows of src
__global__ void k_mean_contig(const float* __restrict__ src, int nn, float inv,
                              float* __restrict__ out) {
    int row = blockIdx.x, col = threadIdx.x << 2;
    const float* p = src + (size_t)row * nn * D + col;
    v4f s = {0.f, 0.f, 0.f, 0.f};
    for (int j = 0; j < nn; ++j) s += *(const v4f*)(p + (size_t)j * D);
    *(v4f*)(out + (size_t)row * D + col) = s * inv;
}

// ---------------- fp32 WMMA GEMM: C[:,coff:coff+128] = op(A) @ W^T + b --------
// A is [M,256] (optionally row-gathered via row_ids), W is [128,256] row-major.
// Block = 256 threads = 8 waves; wave w computes a 32-row x 128-col strip
// (two 16-row M-tiles sharing every B fragment -> halves LDS traffic per FLOP).
// W staged in LDS in two 64KB K-chunks, XOR-swizzled (4-float granule) so the
// 16-lane stride-128 B reads are bank-conflict-free. B fragments for a K-step
// are gathered into a register array first so the 4 ds_load_2addr ops issue
// back-to-back with staggered s_wait_dscnt instead of serializing vs WMMA.
__global__ __launch_bounds__(256)
void k_gemm_wmma(const float* __restrict__ A, const int* __restrict__ row_ids,
                 const float* __restrict__ W, const float* __restrict__ bias,
                 float* __restrict__ C, int col_off, int do_relu) {
    __shared__ float wlds[OUTD * 128];   // 64 KB chunk: W[:, phase*128 .. +128)

    const int tid  = threadIdx.x;
    const int lane = tid & 31;
    const int wv   = tid >> 5;
    const int l15  = lane & 15;
    const int koff = (lane >> 4) << 1;      // 0 for lanes 0-15, 2 for 16-31
    const int tileRow = blockIdx.x * 256 + wv * 32;   // 2 M-tiles per wave
    const int m0  = tileRow + l15;
    const int m1  = m0 + 16;
    const int rid0 = row_ids ? row_ids[m0] : m0;
    const int rid1 = row_ids ? row_ids[m1] : m1;
    const float* arow0 = A + (size_t)rid0 * D + koff;
    const float* arow1 = A + (size_t)rid1 * D + koff;

    const v8f vzero = {0.f, 0.f, 0.f, 0.f, 0.f, 0.f, 0.f, 0.f};
    v8f acc0[8], acc1[8];
#pragma unroll
    for (int n = 0; n < 8; ++n) { acc0[n] = vzero; acc1[n] = vzero; }

    for (int phase = 0; phase < 2; ++phase) {
        // cooperative swizzled load of W[:, phase*128 : phase*128+128]
        __syncthreads();
        for (int idx = tid; idx < OUTD * 128 / 4; idx += 256) {
            int fi  = idx << 2;
            int row = fi >> 7;              // W output-col (0..127)
            int col = fi & 127;             // k within chunk
            float4 w4 = *(const float4*)(W + (size_t)row * D + phase * 128 + col);
            int g = (col >> 2) ^ (row & 15);
            *(float4*)&wlds[row * 128 + (g << 2)] = w4;
        }
        __syncthreads();

        const float* ap0 = arow0 + phase * 128;
        const float* ap1 = arow1 + phase * 128;
#pragma unroll 2
        for (int k0 = 0; k0 < 128; k0 += 4) {
            v2f a0 = *(const v2f*)(ap0 + k0);   // A frag: K=(k0+koff, k0+koff+1)
            v2f a1 = *(const v2f*)(ap1 + k0);

            // gather all 8 B fragments first (4 merged ds_load_2addr ops)
            v2f bfrag[8];
            const int g4 = (((k0 >> 2) ^ l15) << 2) + koff;  // swizzled granule
#pragma unroll
            for (int n = 0; n < 8; ++n)
                bfrag[n] = *(const v2f*)&wlds[(n * 16 + l15) * 128 + g4];

#pragma unroll
            for (int n = 0; n < 8; ++n) {
                acc0[n] = __builtin_amdgcn_wmma_f32_16x16x4_f32(
                    false, a0, false, bfrag[n], (short)0, acc0[n], false, false);
                acc1[n] = __builtin_amdgcn_wmma_f32_16x16x4_f32(
                    false, a1, false, bfrag[n], (short)0, acc1[n], false, false);
            }
        }
    }

    // writeout: v8f element r -> row tile + r (+8 for upper half-wave), col l15
    const int rowAdd = (lane >> 4) << 3;
#pragma unroll
    for (int n = 0; n < 8; ++n) {
        float bv = bias[n * 16 + l15];
        size_t cbase = col_off + n * 16 + l15;
#pragma unroll
        for (int r = 0; r < 8; ++r) {
            float v0 = acc0[n][r] + bv;
            float v1 = acc1[n][r] + bv;
            if (do_relu) { v0 = v0 > 0.f ? v0 : 0.f; v1 = v1 > 0.f ? v1 : 0.f; }
            C[(size_t)(tileRow      + r + rowAdd) * D + cbase] = v0;
            C[(size_t)(tileRow + 16 + r + rowAdd) * D + cbase] = v1;
        }
    }
}

// ------------------------------------ host -----------------------------------
extern "C" void kernel_launch(void* const* d_in, const int* in_sizes, int n_in,
                              void* d_out, int out_size, void* d_ws, size_t ws_size,
                              hipStream_t stream) {
    (void)in_sizes; (void)n_in; (void)out_size; (void)ws_size;
    const int*   ids   = (const int*)d_in[0];
    const int*   adj   = (const int*)d_in[1];
    const float* feats = (const float*)d_in[2];
    const float* W1x   = (const float*)d_in[3];
    const float* b1x   = (const float*)d_in[4];
    const float* W1n   = (const float*)d_in[5];
    const float* b1n   = (const float*)d_in[6];
    const float* W2x   = (const float*)d_in[7];
    const float* b2x   = (const float*)d_in[8];
    const float* W2n   = (const float*)d_in[9];
    const float* b2n   = (const float*)d_in[10];
    float* out = (float*)d_out;

    char* ws = (char*)d_ws;
    size_t o = 0;
    int*   ids1 = (int*)(ws + o);  o += (size_t)N1 * 4;          // 25600 ids
    float* M1   = (float*)(ws + o); o += (size_t)BSZ * D * 4;    // mean(feats[ids1]) per seed
    float* M2   = (float*)(ws + o); o += (size_t)N1  * D * 4;    // mean(feats[ids2]) per ids1
    float* H0   = (float*)(ws + o); o += (size_t)BSZ * D * 4;    // layer1 k=0 out
    float* H1   = (float*)(ws + o); o += (size_t)N1  * D * 4;    // layer1 k=1 out
    float* M3   = (float*)(ws + o);                              // mean(H1) per seed

    // sampling (ids1 only; ids2 folded into k_mean_adj)
    k_sample<<<(N1 + 255) / 256, 256, 0, stream>>>(ids, adj, F1, N1, ids1);

    // neighbor means (64 threads = 2 waves, float4 per lane, 1 row per block)
    k_mean_adj<<<N1,  64, 0, stream>>>(feats, adj, ids1, F2, 1.f / F2, M2);
    k_mean_adj<<<BSZ, 64, 0, stream>>>(feats, adj, ids,  F1, 1.f / F1, M1);

    // layer 1 (relu), concat halves; 256 rows per block
    k_gemm_wmma<<<BSZ / 256, 256, 0, stream>>>(feats, ids,  W1x, b1x, H0, 0,    1);
    k_gemm_wmma<<<BSZ / 256, 256, 0, stream>>>(M1,  nullptr, W1n, b1n, H0, OUTD, 1);
    k_gemm_wmma<<<N1  / 256, 256, 0, stream>>>(feats, ids1, W1x, b1x, H1, 0,    1);
    k_gemm_wmma<<<N1  / 256, 256, 0, stream>>>(M2,  nullptr, W1n, b1n, H1, OUTD, 1);

    // layer 2 (no activation)
    k_mean_contig<<<BSZ, 64, 0, stream>>>(H1, F1, 1.f / F1, M3);
    k_gemm_wmma<<<BSZ / 256, 256, 0, stream>>>(H0, nullptr, W2x, b2x, out, 0,    0);
    k_gemm_wmma<<<BSZ / 256, 256, 0, stream>>>(M3, nullptr, W2n, b2n, out, OUTD, 0);
}